// SequentialLSTM_75196287418593
// MI455X (gfx1250) — compile-verified
//
#include <hip/hip_runtime.h>
#include <hip/hip_bf16.h>

// ---------------- problem constants ----------------
#define SEQ   16384
#define DIN   15
#define HID   1024
#define GATES 4096   // 4*HID
#define HFF   4096

typedef float v2f __attribute__((ext_vector_type(2)));
typedef float v8f __attribute__((ext_vector_type(8)));

// Unconditional load with zero-padding: clamp the index (always in-bounds),
// then select 0 after the load. Compiles to cndmask+load+cndmask, no branches.
__device__ __forceinline__ float ldg_pad(const float* __restrict__ base,
                                         int idx, bool valid) {
    float v = base[valid ? idx : 0];
    return valid ? v : 0.0f;
}

// ---------------------------------------------------------------------------
// Kernel 1: forecasts = (xs @ Wf1.T + bf1) @ Wf2.T + bf2
// One wave per 16-row tile; chained f32 WMMA (16x16x4) with an LDS relayout
// between stage 1 (C-layout) and stage 2 (A-layout). K=15 padded to 16.
// ---------------------------------------------------------------------------
__global__ __launch_bounds__(32) void forecast_kernel(
    const float* __restrict__ xs,    // (SEQ, 15)
    const float* __restrict__ Wf1,   // (4096, 15)
    const float* __restrict__ bf1,   // (4096,)
    const float* __restrict__ Wf2,   // (15, 4096)
    const float* __restrict__ bf2,   // (15,)
    float* __restrict__ fore)        // (SEQ, 15)
{
    __shared__ float ldsT[16 * 16];
    const int lane = threadIdx.x;    // 0..31
    const int lm   = lane & 15;
    const int lh   = lane >> 4;      // 0 or 1
    const int s0   = blockIdx.x * 16;

    // A (xs rows) loaded once. f32 A layout: M = lane%16, K = v + 2*(lane/16).
    v2f a[4];
#pragma unroll
    for (int j = 0; j < 4; ++j) {
        int k = 4 * j + 2 * lh;
        a[j].x = ldg_pad(xs, (s0 + lm) * DIN + k,     k     < DIN);
        a[j].y = ldg_pad(xs, (s0 + lm) * DIN + k + 1, k + 1 < DIN);
    }

    // Stage-2 accumulator init with bf2 (C layout: value depends only on N=lm)
    v8f Fc;
    {
        float c2 = ldg_pad(bf2, lm, lm < DIN);
#pragma unroll
        for (int v = 0; v < 8; ++v) Fc[v] = c2;
    }

    // Row base for the Wf2 fetch (d = lm clamped; masked after load).
    const int d_ok   = (lm < DIN);
    const int d_row  = (d_ok ? lm : 0) * HFF;

    for (int n0 = 0; n0 < HFF; n0 += 16) {
        // ---- stage 1: T = xs @ Wf1.T + bf1 (16x16 tile over cols n0..n0+15)
        v8f T;
        {
            float c1 = bf1[n0 + lm];
#pragma unroll
            for (int v = 0; v < 8; ++v) T[v] = c1;
        }
#pragma unroll
        for (int j = 0; j < 4; ++j) {
            int k = 4 * j + 2 * lh;
            v2f b;
            b.x = ldg_pad(Wf1, (n0 + lm) * DIN + k,     k     < DIN);
            b.y = ldg_pad(Wf1, (n0 + lm) * DIN + k + 1, k + 1 < DIN);
            T = __builtin_amdgcn_wmma_f32_16x16x4_f32(
                    false, a[j], false, b, (short)0, T, false, false);
        }
        // C layout -> row-major LDS
        __syncthreads();
#pragma unroll
        for (int v = 0; v < 8; ++v) ldsT[(v + 8 * lh) * 16 + lm] = T[v];
        __syncthreads();
        // ---- stage 2: Fc += T @ Wf2.T chunk (K = 16 cols of HFF)
#pragma unroll
        for (int j = 0; j < 4; ++j) {
            int k = 4 * j + 2 * lh;
            v2f a2, b2;
            a2.x = ldsT[lm * 16 + k];
            a2.y = ldsT[lm * 16 + k + 1];
            float w0 = Wf2[d_row + n0 + k];
            float w1 = Wf2[d_row + n0 + k + 1];
            b2.x = d_ok ? w0 : 0.0f;
            b2.y = d_ok ? w1 : 0.0f;
            Fc = __builtin_amdgcn_wmma_f32_16x16x4_f32(
                    false, a2, false, b2, (short)0, Fc, false, false);
        }
    }

    if (lm < DIN) {
#pragma unroll
        for (int v = 0; v < 8; ++v) {
            int m = v + 8 * lh;
            fore[(s0 + m) * DIN + lm] = Fc[v];
        }
    }
}

// ---------------------------------------------------------------------------
// Kernel 2: out = A @ W_ih.T + (b_ih + b_hh)   (A is xs or forecasts, K=15)
// One wave per 16x16 output tile; 4x V_WMMA_F32_16X16X4_F32.
// ---------------------------------------------------------------------------
__global__ __launch_bounds__(32) void gates_kernel(
    const float* __restrict__ A,     // (SEQ, 15)
    const float* __restrict__ Wih,   // (4096, 15)
    const float* __restrict__ bih,   // (4096,)
    const float* __restrict__ bhh,   // (4096,)
    float* __restrict__ out)         // (SEQ, 4096)
{
    const int lane = threadIdx.x;
    const int lm   = lane & 15;
    const int lh   = lane >> 4;
    const int s0   = blockIdx.y * 16;
    const int n0   = blockIdx.x * 16;

    v8f C;
    {
        float cinit = bih[n0 + lm] + bhh[n0 + lm];
#pragma unroll
        for (int v = 0; v < 8; ++v) C[v] = cinit;
    }
#pragma unroll
    for (int j = 0; j < 4; ++j) {
        int k = 4 * j + 2 * lh;
        v2f a, b;
        a.x = ldg_pad(A,   (s0 + lm) * DIN + k,     k     < DIN);
        a.y = ldg_pad(A,   (s0 + lm) * DIN + k + 1, k + 1 < DIN);
        b.x = ldg_pad(Wih, (n0 + lm) * DIN + k,     k     < DIN);
        b.y = ldg_pad(Wih, (n0 + lm) * DIN + k + 1, k + 1 < DIN);
        C = __builtin_amdgcn_wmma_f32_16x16x4_f32(
                false, a, false, b, (short)0, C, false, false);
    }
#pragma unroll
    for (int v = 0; v < 8; ++v) {
        int m = v + 8 * lh;
        out[(size_t)(s0 + m) * GATES + n0 + lm] = C[v];
    }
}

// ---------------------------------------------------------------------------
// Kernel 3: the serial scan. One workgroup, 1024 threads (32 wave32 waves on
// one WGP). Thread j owns c[j] and gate lanes j, j+1024, j+2048, j+3072.
// ---------------------------------------------------------------------------
__device__ __forceinline__ float sigmoidf_(float x) {
    return 1.0f / (1.0f + __expf(-x));
}
__device__ __forceinline__ float tanhf_(float x) {
    float e = __expf(-2.0f * fabsf(x));
    float r = (1.0f - e) / (1.0f + e);
    return (x >= 0.0f) ? r : -r;
}

__device__ __forceinline__ float block_reduce_1024(float v, float* red,
                                                   float* hres, int lane,
                                                   int wid) {
#pragma unroll
    for (int off = 16; off > 0; off >>= 1) v += __shfl_xor(v, off, 32);
    if (lane == 0) red[wid] = v;
    __syncthreads();
    if (wid == 0) {
        float x = red[lane];
#pragma unroll
        for (int off = 16; off > 0; off >>= 1) x += __shfl_xor(x, off, 32);
        if (lane == 0) hres[0] = x;
    }
    __syncthreads();
    return hres[0];
}

template <bool STREAM>
__global__ __launch_bounds__(1024, 1) void scan_kernel(
    const float* __restrict__ xs,    // (SEQ, 15)
    const float* __restrict__ fore,  // (SEQ, 15)
    const float* __restrict__ gx,    // (SEQ, 4096) if STREAM
    const float* __restrict__ gf,    // (SEQ, 4096) if STREAM
    const float* __restrict__ Wih,   // (4096, 15)  if !STREAM
    const float* __restrict__ bih,
    const float* __restrict__ bhh,
    const float* __restrict__ whh,   // (4096,)
    const float* __restrict__ whr,   // (1024,)
    float* __restrict__ prog,        // (SEQ,)
    float* __restrict__ fprog)       // (SEQ,)
{
    __shared__ float sred[2][32];
    __shared__ float shres[2];

    const int j    = threadIdx.x;    // 0..1023 == c index
    const int lane = j & 31;
    const int wid  = j >> 5;

    const float wi = whh[j];
    const float wf = whh[j + HID];
    const float wg = whh[j + 2 * HID];
    const float wo = whh[j + 3 * HID];
    const float wr = whr[j];

    float bi_ = 0, bf_ = 0, bg_ = 0, bo_ = 0;
    float Wri[DIN], Wrf[DIN], Wrg[DIN], Wro[DIN];
    if constexpr (!STREAM) {
        bi_ = bih[j]           + bhh[j];
        bf_ = bih[j + HID]     + bhh[j + HID];
        bg_ = bih[j + 2 * HID] + bhh[j + 2 * HID];
        bo_ = bih[j + 3 * HID] + bhh[j + 3 * HID];
#pragma unroll
        for (int k = 0; k < DIN; ++k) {
            Wri[k] = Wih[(size_t)j * DIN + k];
            Wrf[k] = Wih[(size_t)(j + HID) * DIN + k];
            Wrg[k] = Wih[(size_t)(j + 2 * HID) * DIN + k];
            Wro[k] = Wih[(size_t)(j + 3 * HID) * DIN + k];
        }
    }

    float c = 0.0f;
    float h = 0.0f;

    for (int t = 0; t < SEQ; ++t) {
        float pi, pf, pg, po, qi, qf, qg, qo;
        if constexpr (STREAM) {
            const float* r1 = gx + (size_t)t * GATES;
            const float* r2 = gf + (size_t)t * GATES;
            pi = r1[j];           pf = r1[j + HID];
            pg = r1[j + 2 * HID]; po = r1[j + 3 * HID];
            qi = r2[j];           qf = r2[j + HID];
            qg = r2[j + 2 * HID]; qo = r2[j + 3 * HID];
            if (t + 8 < SEQ) {   // pull future rows toward L2/L0 off the path
                const float* p1 = gx + (size_t)(t + 8) * GATES;
                const float* p2 = gf + (size_t)(t + 8) * GATES;
#pragma unroll
                for (int u = 0; u < 4; ++u) {
                    __builtin_prefetch(p1 + j + u * HID, 0, 1);
                    __builtin_prefetch(p2 + j + u * HID, 0, 1);
                }
            }
        } else {
            pi = bi_; pf = bf_; pg = bg_; po = bo_;
            qi = bi_; qf = bf_; qg = bg_; qo = bo_;
#pragma unroll
            for (int k = 0; k < DIN; ++k) {
                float xv = xs[t * DIN + k];      // uniform -> scalar load
                float fv = fore[t * DIN + k];
                pi += Wri[k] * xv; pf += Wrf[k] * xv;
                pg += Wrg[k] * xv; po += Wro[k] * xv;
                qi += Wri[k] * fv; qf += Wrf[k] * fv;
                qg += Wrg[k] * fv; qo += Wro[k] * fv;
            }
        }

        // ---- cell 1 (updates carry c, produces scalar h1 = prog[t])
        float ai = sigmoidf_(pi + h * wi);
        float af = sigmoidf_(pf + h * wf);
        float ag = tanhf_(pg + h * wg);
        float ao = sigmoidf_(po + h * wo);
        c = af * c + ai * ag;
        float part1 = ao * tanhf_(c) * wr;
        float h1 = block_reduce_1024(part1, sred[0], &shres[0], lane, wid);

        // ---- cell 2 (uses (h1, c1); its c is discarded) -> h2 = fprog[t]
        float ci = sigmoidf_(qi + h1 * wi);
        float cf = sigmoidf_(qf + h1 * wf);
        float cg = tanhf_(qg + h1 * wg);
        float co = sigmoidf_(qo + h1 * wo);
        float c2 = cf * c + ci * cg;
        float part2 = co * tanhf_(c2) * wr;
        float h2 = block_reduce_1024(part2, sred[1], &shres[1], lane, wid);

        h = h1;
        if (j == 0) {
            prog[t]  = h1;
            fprog[t] = h2;
        }
    }
}

// ---------------------------------------------------------------------------
extern "C" void kernel_launch(void* const* d_in, const int* in_sizes, int n_in,
                              void* d_out, int out_size, void* d_ws,
                              size_t ws_size, hipStream_t stream) {
    const float* x    = (const float*)d_in[0];  // (1, SEQ, 15)
    const float* W_ih = (const float*)d_in[1];  // (4096, 15)
    const float* W_hh = (const float*)d_in[2];  // (4096, 1) -> whh
    const float* b_ih = (const float*)d_in[3];  // (4096,)
    const float* b_hh = (const float*)d_in[4];  // (4096,)
    const float* W_hr = (const float*)d_in[5];  // (1, 1024) -> whr
    const float* Wf1  = (const float*)d_in[6];  // (4096, 15)
    const float* bf1  = (const float*)d_in[7];  // (4096,)
    const float* Wf2  = (const float*)d_in[8];  // (15, 4096)
    const float* bf2  = (const float*)d_in[9];  // (15,)

    float* out   = (float*)d_out;
    float* prog  = out;                 // (SEQ,)
    float* fprog = out + SEQ;           // (SEQ,)
    float* fore  = out + 2 * SEQ;       // (SEQ, 15)

    // 1) forecasts via chained WMMA GEMMs
    forecast_kernel<<<SEQ / 16, 32, 0, stream>>>(x, Wf1, bf1, Wf2, bf2, fore);

    const size_t need = 2ull * SEQ * GATES * sizeof(float);  // gx + gf
    if (ws_size >= need) {
        float* gx = (float*)d_ws;
        float* gf = gx + (size_t)SEQ * GATES;
        dim3 grid(GATES / 16, SEQ / 16);
        gates_kernel<<<grid, 32, 0, stream>>>(x, W_ih, b_ih, b_hh, gx);
        gates_kernel<<<grid, 32, 0, stream>>>(fore, W_ih, b_ih, b_hh, gf);
        scan_kernel<true><<<1, 1024, 0, stream>>>(
            x, fore, gx, gf, W_ih, b_ih, b_hh, W_hh, W_hr, prog, fprog);
    } else {
        scan_kernel<false><<<1, 1024, 0, stream>>>(
            x, fore, nullptr, nullptr, W_ih, b_ih, b_hh, W_hh, W_hr, prog,
            fprog);
    }
}